// nSGC_26680336843068
// MI455X (gfx1250) — compile-verified
//
#include <hip/hip_runtime.h>
#include <hip/hip_bf16.h>

typedef __attribute__((ext_vector_type(2))) float v2f;
typedef __attribute__((ext_vector_type(8))) float v8f;

#define HID  512
#define FEAT 2560   // HID * (K_HOPS + 1)

// -------------------- degree + symmetric norm --------------------
__global__ void deg_kernel(const int* __restrict__ dst, float* __restrict__ deg, int E) {
  int e = blockIdx.x * blockDim.x + threadIdx.x;
  if (e < E) atomicAdd(&deg[dst[e]], 1.0f);
}

__global__ void norm_kernel(float* __restrict__ deg, int n) {
  int i = blockIdx.x * blockDim.x + threadIdx.x;
  if (i < n) {
    float d = deg[i];
    deg[i] = rsqrtf(d < 1.0f ? 1.0f : d);   // clamp(deg,1)^-0.5
  }
}

// -------------------- fp32 WMMA GEMM: C = A*B (+bias) --------------------
// Block: 256 threads = 8 waves. Wave w -> rows [bx*256 + w*32, +32) as two
// 16-row A fragments, cols [by*128, +128) as eight 16x16 WMMA tiles.
// K steps of 4, 2-stage software pipeline (fetch k+4 while WMMA on k).

struct Frag {
  v2f a0, a1;   // two 16x4 A fragments (rows r..r+15, r+16..r+31)
  v2f b[8];     // eight 4x16 B fragments (col tiles)
};

__device__ __forceinline__ void load_frag(const float* __restrict__ Arow0,
                                          const float* __restrict__ Arow1,
                                          const float* __restrict__ Bbase,
                                          int k, int kq, int ldb, Frag& f) {
  f.a0 = *(const v2f*)(Arow0 + k);
  f.a1 = *(const v2f*)(Arow1 + k);
  const float* Bk = Bbase + (size_t)(k + kq) * ldb;
  #pragma unroll
  for (int t = 0; t < 8; ++t) {
    v2f b;
    b[0] = Bk[t * 16];         // B[k+kq  ][col]
    b[1] = Bk[t * 16 + ldb];   // B[k+kq+1][col]
    f.b[t] = b;
  }
}

__global__ void __launch_bounds__(256)
gemm_wmma_f32(const float* __restrict__ A, const float* __restrict__ B,
              const float* __restrict__ bias, float* __restrict__ C,
              int M, int K, int lda, int ldb, int ldc) {
  const int lane = threadIdx.x & 31;
  const int wave = threadIdx.x >> 5;
  const int row0 = blockIdx.x * 256 + wave * 32;
  const int col0 = blockIdx.y * 128;
  if (row0 >= M) return;

  const int m  = lane & 15;          // row (A) / col (B,C) within 16-tile
  const int kq = (lane >> 4) << 1;   // lanes 0-15 -> K=0,1 ; lanes 16-31 -> K=2,3

  const float* Arow0 = A + (size_t)(row0 + m) * lda + kq;
  const float* Arow1 = A + (size_t)(row0 + 16 + m) * lda + kq;
  const float* Bbase = B + col0 + m;

  v8f acc[2][8] = {};

  Frag cur, nxt;
  load_frag(Arow0, Arow1, Bbase, 0, kq, ldb, cur);

  int k = 0;
  for (; k + 4 < K; k += 4) {
    load_frag(Arow0, Arow1, Bbase, k + 4, kq, ldb, nxt);
    #pragma unroll
    for (int t = 0; t < 8; ++t) {
      acc[0][t] = __builtin_amdgcn_wmma_f32_16x16x4_f32(
          false, cur.a0, false, cur.b[t], (short)0, acc[0][t], false, false);
      acc[1][t] = __builtin_amdgcn_wmma_f32_16x16x4_f32(
          false, cur.a1, false, cur.b[t], (short)0, acc[1][t], false, false);
    }
    cur = nxt;
  }
  #pragma unroll
  for (int t = 0; t < 8; ++t) {     // peeled last step
    acc[0][t] = __builtin_amdgcn_wmma_f32_16x16x4_f32(
        false, cur.a0, false, cur.b[t], (short)0, acc[0][t], false, false);
    acc[1][t] = __builtin_amdgcn_wmma_f32_16x16x4_f32(
        false, cur.a1, false, cur.b[t], (short)0, acc[1][t], false, false);
  }

  // C/D layout: lanes 0-15 -> M=r, lanes 16-31 -> M=r+8; N = lane&15
  const int mb = (lane >> 4) * 8;
  #pragma unroll
  for (int g = 0; g < 2; ++g) {
    #pragma unroll
    for (int t = 0; t < 8; ++t) {
      int colg = col0 + t * 16 + m;
      float bv = bias ? bias[colg] : 0.0f;
      #pragma unroll
      for (int r = 0; r < 8; ++r) {
        C[(size_t)(row0 + g * 16 + mb + r) * ldc + colg] = acc[g][t][r] + bv;
      }
    }
  }
}

// -------------------- one propagation hop: gather * w -> atomic scatter ----
// 128 threads per edge, float4 per thread (512 features).
__global__ void propagate_kernel(const int* __restrict__ src, const int* __restrict__ dst,
                                 const float* __restrict__ nrm, float* __restrict__ feat,
                                 int colx, int coly, int E) {
  int t = blockIdx.x * blockDim.x + threadIdx.x;
  int e = t >> 7;
  if (e >= E) return;
  int f = (t & 127) << 2;
  int s = src[e], d = dst[e];
  float w = nrm[s] * nrm[d];
  const float4 x = *(const float4*)(feat + (size_t)s * FEAT + colx + f);
  float* o = feat + (size_t)d * FEAT + coly + f;
  atomicAdd(o + 0, x.x * w);
  atomicAdd(o + 1, x.y * w);
  atomicAdd(o + 2, x.z * w);
  atomicAdd(o + 3, x.w * w);
}

// -------------------- per-node dots with Wp[0:512], Wp[512:1024] -----------
__global__ void node_dots_kernel(const float* __restrict__ h, const float* __restrict__ Wp,
                                 float* __restrict__ sd, float* __restrict__ sm, int n) {
  int gid  = blockIdx.x * blockDim.x + threadIdx.x;
  int node = gid >> 5;
  int lane = gid & 31;
  if (node >= n) return;
  const float* hr = h + (size_t)node * HID;
  float a = 0.f, b = 0.f;
  for (int i = lane; i < HID; i += 32) {
    float v = hr[i];
    a += v * Wp[i];
    b += v * Wp[HID + i];
  }
  for (int off = 16; off > 0; off >>= 1) {
    a += __shfl_xor(a, off, 32);
    b += __shfl_xor(b, off, 32);
  }
  if (lane == 0) { sd[node] = a; sm[node] = b; }
}

// -------------------- per-pair cross term + sigmoid ------------------------
__global__ void pair_kernel(const float* __restrict__ h, const float* __restrict__ Wp,
                            const float* __restrict__ sd, const float* __restrict__ sm,
                            const int* __restrict__ dis, const int* __restrict__ mir,
                            const float* __restrict__ bp, float* __restrict__ out, int P) {
  int gid  = blockIdx.x * blockDim.x + threadIdx.x;
  int p    = gid >> 5;
  int lane = gid & 31;
  if (p >= P) return;
  int dn = dis[p], mn = mir[p];   // mirnas already offset by N_D in inputs
  const float* hd = h + (size_t)dn * HID;
  const float* hm = h + (size_t)mn * HID;
  float c = 0.f;
  for (int i = lane; i < HID; i += 32)
    c += hd[i] * hm[i] * Wp[2 * HID + i];
  for (int off = 16; off > 0; off >>= 1) c += __shfl_xor(c, off, 32);
  if (lane == 0) {
    float z = sd[dn] + sm[mn] + c + bp[0];
    out[p] = 1.0f / (1.0f + __expf(-z));
  }
}

extern "C" void kernel_launch(void* const* d_in, const int* in_sizes, int n_in,
                              void* d_out, int out_size, void* d_ws, size_t ws_size,
                              hipStream_t stream) {
  const float* d_sim = (const float*)d_in[0];
  const float* m_sim = (const float*)d_in[1];
  const int*   src   = (const int*)d_in[2];
  const int*   dst   = (const int*)d_in[3];
  const int*   dis   = (const int*)d_in[4];
  const int*   mir   = (const int*)d_in[5];
  const float* Wd    = (const float*)d_in[6];
  const float* Wm    = (const float*)d_in[7];
  const float* Wf    = (const float*)d_in[8];
  const float* bf    = (const float*)d_in[9];
  const float* Wp    = (const float*)d_in[10];
  const float* bp    = (const float*)d_in[11];

  const int N_D = 4096, N_M = 8192, Nn = N_D + N_M;
  const int E = in_sizes[2];
  const int P = in_sizes[4];

  // Workspace layout (fp32): feat0[N,2560] | h[N,512] | norm[N] | sd[N] | sm[N]
  float* feat0 = (float*)d_ws;
  float* hbuf  = feat0 + (size_t)Nn * FEAT;
  float* nrm   = hbuf + (size_t)Nn * HID;
  float* sd    = nrm + Nn;
  float* sm    = sd + Nn;

  hipMemsetAsync(feat0, 0, (size_t)Nn * FEAT * sizeof(float), stream);
  hipMemsetAsync(nrm, 0, (size_t)Nn * sizeof(float), stream);

  deg_kernel <<<(E + 255) / 256, 256, 0, stream>>>(dst, nrm, E);
  norm_kernel<<<(Nn + 255) / 256, 256, 0, stream>>>(nrm, Nn);

  // X = [d_sim@Wd ; m_sim@Wm] written into feat0 columns [0,512)
  gemm_wmma_f32<<<dim3(N_D / 256, HID / 128), 256, 0, stream>>>(
      d_sim, Wd, nullptr, feat0, N_D, N_D, N_D, HID, FEAT);
  gemm_wmma_f32<<<dim3(N_M / 256, HID / 128), 256, 0, stream>>>(
      m_sim, Wm, nullptr, feat0 + (size_t)N_D * FEAT, N_M, N_M, N_M, HID, FEAT);

  // 4 hops: read cols [hop*512, +512), atomically accumulate into next 512 cols
  int pthreads = E * 128;
  for (int hop = 0; hop < 4; ++hop) {
    propagate_kernel<<<(pthreads + 255) / 256, 256, 0, stream>>>(
        src, dst, nrm, feat0, hop * HID, (hop + 1) * HID, E);
  }

  // h = feat0 @ Wf + bf
  gemm_wmma_f32<<<dim3(Nn / 256, HID / 128), 256, 0, stream>>>(
      feat0, Wf, bf, hbuf, Nn, FEAT, FEAT, HID, HID);

  node_dots_kernel<<<(Nn * 32 + 255) / 256, 256, 0, stream>>>(hbuf, Wp, sd, sm, Nn);
  pair_kernel<<<(P * 32 + 255) / 256, 256, 0, stream>>>(hbuf, Wp, sd, sm, dis, mir, bp,
                                                        (float*)d_out, P);
}